// RWKV_x_33139967656544
// MI455X (gfx1250) — compile-verified
//
#include <hip/hip_runtime.h>
#include <hip/hip_bf16.h>

typedef float v2f  __attribute__((ext_vector_type(2)));
typedef float v8f  __attribute__((ext_vector_type(8)));
typedef unsigned int u32x4 __attribute__((ext_vector_type(4)));
typedef int   i32x4 __attribute__((ext_vector_type(4)));
typedef int   i32x8 __attribute__((ext_vector_type(8)));

#define Bdim 4
#define Tdim 2048
#define Hdim 16
#define Ndim 64
#define CL   16            // chunk length
#define NCHUNK (Tdim / CL)
#define NTHREADS 128       // 4 wave32

// ---- Tensor Data Mover: load a 16x64 f32 tile (row stride Hdim*Ndim elems)
// from global into contiguous LDS.  D# packing per CDNA5 ISA §8.3-8.5.
__device__ __forceinline__ void tdm_load_tile(const float* gptr, unsigned lds_off) {
    unsigned long long ga = (unsigned long long)(uintptr_t)gptr;
    u32x4 g0;
    g0[0] = 1u;                                       // count=1, user desc
    g0[1] = lds_off;                                  // lds_addr (bytes)
    g0[2] = (unsigned)(ga & 0xffffffffull);           // global_addr[31:0]
    g0[3] = (unsigned)((ga >> 32) & 0x01ffffffull)    // global_addr[56:32]
          | (2u << 30);                               // type=2 (image)
    i32x8 g1;
    g1[0] = (int)(2u << 16);                          // data_size=2 (4 bytes)
    g1[1] = (int)((unsigned)Ndim << 16);              // tensor_dim0 = 64
    g1[2] = 0;                                        // tensor_dim1[15:0] = 0
    g1[3] = (int)(0x0100u | ((unsigned)Ndim << 16));  // tensor_dim1=1<<24, tile_dim0=64
    g1[4] = CL;                                       // tile_dim1 = 16
    g1[5] = Hdim * Ndim;                              // tensor_dim0_stride = 1024
    g1[6] = 0;
    g1[7] = 0;
    i32x4 z4 = {0, 0, 0, 0};
#if defined(__clang_major__) && (__clang_major__ >= 23)
    i32x8 z8 = {0, 0, 0, 0, 0, 0, 0, 0};
    __builtin_amdgcn_tensor_load_to_lds(g0, g1, z4, z4, z8, 0);
#else
    __builtin_amdgcn_tensor_load_to_lds(g0, g1, z4, z4, 0);
#endif
}

__device__ __forceinline__ unsigned lds_off_of(const void* p) {
    return (unsigned)(uintptr_t)p;    // LDS aperture: addr[31:0] = LDS byte offset
}

__global__ __launch_bounds__(NTHREADS)
void rwkv6_chunked_wmma(const float* __restrict__ r,
                        const float* __restrict__ k,
                        const float* __restrict__ v,
                        const float* __restrict__ w,
                        const float* __restrict__ u,
                        float* __restrict__ o)
{
    const int bh   = blockIdx.x;
    const int b    = bh / Hdim;
    const int h    = bh % Hdim;
    const int tid  = threadIdx.x;
    const int wave = tid >> 5;       // 0..3
    const int lane = tid & 31;
    const int half = lane >> 4;
    const int l16  = lane & 15;

    // ---- LDS ----
    __shared__ alignas(16) float S[Ndim][Ndim + 1];     // state (col slices wave-private)
    __shared__ alignas(16) float bufr[2][CL][Ndim];     // double-buffered chunk inputs (TDM dst)
    __shared__ alignas(16) float bufk[2][CL][Ndim];
    __shared__ alignas(16) float bufv[2][CL][Ndim];
    __shared__ alignas(16) float bufw[2][CL][Ndim];
    __shared__ alignas(16) float rq[CL][Ndim];          // r * exp(cw[t])
    __shared__ alignas(16) float kh[CL][Ndim];          // k * exp(cw[CL] - cw[s+1])
    __shared__ alignas(16) float cw[CL + 1][Ndim];      // cumulative log-decay (<=0)
    __shared__ float pt[Ndim];                          // exp(cw[CL])
    __shared__ float Amat[CL][CL + 1];                  // masked intra-chunk scores
    __shared__ float us[Ndim];

    if (tid < Ndim) us[tid] = u[h * Ndim + tid];
    for (int idx = tid; idx < Ndim * Ndim; idx += NTHREADS)
        S[idx >> 6][idx & 63] = 0.0f;

    const int col = 16 * wave + l16;          // this lane's N-column (B/C frags)
    const size_t base0 = (((size_t)b * Tdim) * Hdim + h) * Ndim;

    // ---- prologue: TDM-prefetch chunk 0 (one tensor per wave) ----
    {
        const size_t off = base0;             // t0 = 0
        if      (wave == 0) tdm_load_tile(r + off, lds_off_of(&bufr[0][0][0]));
        else if (wave == 1) tdm_load_tile(k + off, lds_off_of(&bufk[0][0][0]));
        else if (wave == 2) tdm_load_tile(v + off, lds_off_of(&bufv[0][0][0]));
        else                tdm_load_tile(w + off, lds_off_of(&bufw[0][0][0]));
    }

    for (int c = 0; c < NCHUNK; ++c) {
        const int t0  = c * CL;
        const int cur = c & 1;
        const int nxt = cur ^ 1;

        __builtin_amdgcn_s_wait_tensorcnt(0);  // this wave's tile for chunk c landed
        __syncthreads();                       // all tiles visible; prev phase4 done

        // ---- issue TDM for chunk c+1 while we compute on chunk c ----
        if (c + 1 < NCHUNK) {
            const size_t off = base0 + (size_t)(t0 + CL) * (Hdim * Ndim);
            if      (wave == 0) tdm_load_tile(r + off, lds_off_of(&bufr[nxt][0][0]));
            else if (wave == 1) tdm_load_tile(k + off, lds_off_of(&bufk[nxt][0][0]));
            else if (wave == 2) tdm_load_tile(v + off, lds_off_of(&bufv[nxt][0][0]));
            else                tdm_load_tile(w + off, lds_off_of(&bufw[nxt][0][0]));
        }

        const float (*rs)[Ndim] = bufr[cur];
        const float (*ks)[Ndim] = bufk[cur];
        const float (*vs)[Ndim] = bufv[cur];
        const float (*ws)[Ndim] = bufw[cur];

        // ---- phase 2: per-channel prefix of log-decay (register chain) ----
        if (tid < Ndim) {
            float e[CL];
#pragma unroll
            for (int t = 0; t < CL; ++t) e[t] = __expf(ws[t][tid]);
            float acc = 0.0f;
            cw[0][tid] = 0.0f;
#pragma unroll
            for (int t = 0; t < CL; ++t) { acc -= e[t]; cw[t + 1][tid] = acc; }
            pt[tid] = __expf(acc);
        }
        __syncthreads();

        // ---- phase 3a: decay-scaled operands (exponents <= 0 -> stable) ----
        for (int idx = tid; idx < CL * Ndim; idx += NTHREADS) {
            const int t = idx >> 6, i = idx & 63;
            rq[t][i] = rs[t][i] * __expf(cw[t][i]);
            kh[t][i] = ks[t][i] * __expf(cw[CL][i] - cw[t + 1][i]);
        }
        // ---- phase 3b: exact intra-chunk scores, b128 LDS reads ----
        for (int idx = tid; idx < CL * CL; idx += NTHREADS) {
            const int t = idx >> 4, s = idx & 15;
            float a = 0.0f;
            if (s < t) {
                for (int i = 0; i < Ndim; i += 4) {
                    const float4 rt = *(const float4*)&rs[t][i];
                    const float4 kt = *(const float4*)&ks[s][i];
                    const float4 ct = *(const float4*)&cw[t][i];
                    const float4 cs = *(const float4*)&cw[s + 1][i];
                    a += rt.x * kt.x * __expf(ct.x - cs.x);
                    a += rt.y * kt.y * __expf(ct.y - cs.y);
                    a += rt.z * kt.z * __expf(ct.z - cs.z);
                    a += rt.w * kt.w * __expf(ct.w - cs.w);
                }
            } else if (s == t) {
                for (int i = 0; i < Ndim; i += 4) {
                    const float4 rt = *(const float4*)&rs[t][i];
                    const float4 kt = *(const float4*)&ks[t][i];
                    const float4 uu = *(const float4*)&us[i];
                    a += rt.x * uu.x * kt.x + rt.y * uu.y * kt.y
                       + rt.z * uu.z * kt.z + rt.w * uu.w * kt.w;
                }
            }
            Amat[t][s] = a;
        }
        __syncthreads();

        // ---- phase 4: WMMA (EXEC all ones, wave32) ----
        // A-frag (16x4 f32): M = lane%16, K = 2*half + comp
        // B-frag (4x16 f32): N = lane%16, K = 2*half + comp
        // C/D v8f: row = vgpr + 8*half, col = lane%16
        v8f oacc = {0.f, 0.f, 0.f, 0.f, 0.f, 0.f, 0.f, 0.f};

        // O_inter = RQ @ S   (K = 64 channels -> 16 wmma)
#pragma unroll
        for (int kk = 0; kk < 16; ++kk) {
            const int kb = 4 * kk + 2 * half;
            v2f a;  a.x = rq[l16][kb];  a.y = rq[l16][kb + 1];
            v2f bb; bb.x = S[kb][col];  bb.y = S[kb + 1][col];
            oacc = __builtin_amdgcn_wmma_f32_16x16x4_f32(
                false, a, false, bb, (short)0, oacc, false, false);
        }
        // O += A @ V   (K = 16 chunk steps -> 4 wmma)
#pragma unroll
        for (int kk = 0; kk < 4; ++kk) {
            const int kb = 4 * kk + 2 * half;
            v2f a;  a.x = Amat[l16][kb]; a.y = Amat[l16][kb + 1];
            v2f bb; bb.x = vs[kb][col];  bb.y = vs[kb + 1][col];
            oacc = __builtin_amdgcn_wmma_f32_16x16x4_f32(
                false, a, false, bb, (short)0, oacc, false, false);
        }
        // write output tile
#pragma unroll
        for (int j = 0; j < 8; ++j) {
            const int m = 8 * half + j;
            const size_t g = (((size_t)b * Tdim + (t0 + m)) * Hdim + h) * Ndim + col;
            o[g] = oacc[j];
        }

        // state update: S = diag(pt) * S + KH^T @ V  (4 row-tiles x 4 wmma)
#pragma unroll
        for (int gt = 0; gt < 4; ++gt) {
            v8f sc;
#pragma unroll
            for (int j = 0; j < 8; ++j) {
                const int m = 16 * gt + 8 * half + j;
                sc[j] = S[m][col] * pt[m];
            }
#pragma unroll
            for (int kk = 0; kk < 4; ++kk) {
                const int kb = 4 * kk + 2 * half;
                const int i = 16 * gt + l16;     // A-matrix row = channel i
                v2f a;  a.x = kh[kb][i];    a.y = kh[kb + 1][i];
                v2f bb; bb.x = vs[kb][col]; bb.y = vs[kb + 1][col];
                sc = __builtin_amdgcn_wmma_f32_16x16x4_f32(
                    false, a, false, bb, (short)0, sc, false, false);
            }
#pragma unroll
            for (int j = 0; j < 8; ++j) {
                const int m = 16 * gt + 8 * half + j;
                S[m][col] = sc[j];
            }
        }
    }
}

extern "C" void kernel_launch(void* const* d_in, const int* in_sizes, int n_in,
                              void* d_out, int out_size, void* d_ws, size_t ws_size,
                              hipStream_t stream) {
    (void)in_sizes; (void)n_in; (void)out_size; (void)d_ws; (void)ws_size;
    const float* r = (const float*)d_in[0];
    const float* k = (const float*)d_in[1];
    const float* v = (const float*)d_in[2];
    const float* w = (const float*)d_in[3];
    const float* u = (const float*)d_in[4];
    float* o = (float*)d_out;

    dim3 grid(Bdim * Hdim);       // one block per (batch, head)
    dim3 block(NTHREADS);         // 4 wave32
    rwkv6_chunked_wmma<<<grid, block, 0, stream>>>(r, k, v, w, u, o);
}